// PVConv_16217796510258
// MI455X (gfx1250) — compile-verified
//
#include <hip/hip_runtime.h>
#include <hip/hip_bf16.h>

typedef float v2f __attribute__((ext_vector_type(2)));
typedef float v8f __attribute__((ext_vector_type(8)));

#define BATCH 8
#define CH 64
#define NPTS 16384
#define RR 32
#define R3 32768            // 32^3
#define ALPHA 0.1f
#define BN_EPS 1e-4f
#define VOX_EPS 1e-6f

__device__ __forceinline__ float lrelu(float x) { return x >= 0.f ? x : ALPHA * x; }

// ---------------- coords mean per batch ----------------
__global__ __launch_bounds__(256) void k_coord_mean(const float* __restrict__ coords,
                                                    float* __restrict__ bmean) {
    __shared__ float red[256];
    int b = blockIdx.x;
    for (int d = 0; d < 3; d++) {
        const float* base = coords + ((size_t)b * 3 + d) * NPTS;
        float s = 0.f;
        for (int n = threadIdx.x; n < NPTS; n += 256) s += base[n];
        red[threadIdx.x] = s; __syncthreads();
        for (int st = 128; st > 0; st >>= 1) {
            if (threadIdx.x < st) red[threadIdx.x] += red[threadIdx.x + st];
            __syncthreads();
        }
        if (threadIdx.x == 0) bmean[b * 3 + d] = red[0] / (float)NPTS;
        __syncthreads();
    }
}

// ---------------- max centered norm per batch ----------------
__global__ __launch_bounds__(256) void k_coord_maxnorm(const float* __restrict__ coords,
                                                       const float* __restrict__ bmean,
                                                       float* __restrict__ bmax) {
    __shared__ float red[256];
    int b = blockIdx.x;
    float m0 = bmean[b * 3], m1 = bmean[b * 3 + 1], m2 = bmean[b * 3 + 2];
    const float* cb = coords + (size_t)b * 3 * NPTS;
    float mx = 0.f;
    for (int n = threadIdx.x; n < NPTS; n += 256) {
        float x = cb[n] - m0;
        float y = cb[NPTS + n] - m1;
        float z = cb[2 * NPTS + n] - m2;
        mx = fmaxf(mx, sqrtf(x * x + y * y + z * z));
    }
    red[threadIdx.x] = mx; __syncthreads();
    for (int st = 128; st > 0; st >>= 1) {
        if (threadIdx.x < st) red[threadIdx.x] = fmaxf(red[threadIdx.x], red[threadIdx.x + st]);
        __syncthreads();
    }
    if (threadIdx.x == 0) bmax[b] = red[0];
}

// ---------------- voxelize: norm coords + scatter-add ----------------
// grid layout: [B][R^3][C] channel-last. cnt: [B][R^3]
__global__ __launch_bounds__(256) void k_voxelize(const float* __restrict__ coords,
                                                  const float* __restrict__ feat,
                                                  const float* __restrict__ bmean,
                                                  const float* __restrict__ bmax,
                                                  float* __restrict__ grid,
                                                  float* __restrict__ cnt,
                                                  float* __restrict__ nrm) {
    int i = blockIdx.x * 256 + threadIdx.x;
    int b = i >> 14;           // / NPTS
    int n = i & (NPTS - 1);
    float inv = 1.f / (2.f * bmax[b] + VOX_EPS);
    int vox[3];
    for (int d = 0; d < 3; d++) {
        float c = coords[((size_t)b * 3 + d) * NPTS + n];
        float nc = (c - bmean[b * 3 + d]) * inv + 0.5f;
        nc = fminf(fmaxf(nc * (float)RR, 0.f), (float)(RR - 1));
        nrm[((size_t)b * 3 + d) * NPTS + n] = nc;
        vox[d] = (int)rintf(nc);   // round-half-even, matches jnp.round
    }
    int idx = (vox[0] * RR + vox[1]) * RR + vox[2];
    atomicAdd(&cnt[b * R3 + idx], 1.f);
    float* gb = grid + ((size_t)(b * R3 + idx)) * CH;
    const float* fb = feat + ((size_t)b * CH) * NPTS + n;
    for (int c = 0; c < CH; c++) atomicAdd(&gb[c], fb[(size_t)c * NPTS]);
}

// ---------------- grid = sums / max(cnt,1) ----------------
__global__ __launch_bounds__(256) void k_grid_norm(float* __restrict__ grid,
                                                   const float* __restrict__ cnt) {
    size_t i = (size_t)blockIdx.x * 256 + threadIdx.x;  // over B*R3*CH
    grid[i] = grid[i] / fmaxf(cnt[i >> 6], 1.f);
}

// ---------------- implicit-GEMM 3x3x3 conv via V_WMMA_F32_16X16X4_F32 --------
// in/out: [B][R^3][CH] channel-last. w: [3][3][3][Cin][Cout]. Optional fused
// input BN+LeakyReLU via per-channel scale/shift (nullptr = raw input).
// Block: 128 consecutive spatial positions (4 y-rows of 32 x, same z) x 64 Cout.
// 8 waves; wave w owns M rows [w*16, w*16+16), all 64 Cout (4 accumulators).
// A tile [m][ci], even stride 68 -> aligned ds_load_b64 of the (K,K+1) pair;
// two-phase b64 banks: low words 4j+ka / 4j+ka+2 (halves disjoint mod 4).
// B tile K-pair interleaved [kpair][co][2], stride 160 -> aligned ds_load_b64;
// row offset 160 mod 64 = 32, so hi=1 half uses banks 32..63: conflict-free.
#define A_STRIDE 68
#define BP_STRIDE 160
__global__ __launch_bounds__(256) void k_conv_wmma(const float* __restrict__ in,
                                                   float* __restrict__ out,
                                                   const float* __restrict__ w,
                                                   const float* __restrict__ bias,
                                                   const float* __restrict__ scale,
                                                   const float* __restrict__ shift) {
    __shared__ float Alds[128 * A_STRIDE];   // [m][ci]
    __shared__ float Blds[32 * BP_STRIDE];   // [kpair][co*2 + (k&1)]
    int tid = threadIdx.x;
    int blk = blockIdx.x;
    int b   = blk >> 8;              // 256 m-blocks per batch (32768/128)
    int sp0 = (blk & 255) * 128;
    int z0  = sp0 >> 10;
    int y0  = (sp0 >> 5) & 31;
    int lane = tid & 31;
    int wave = tid >> 5;
    int lo = lane & 15, hi = lane >> 4;

    v8f acc[4];
    const v8f vzero = {0.f, 0.f, 0.f, 0.f, 0.f, 0.f, 0.f, 0.f};
#pragma unroll
    for (int i = 0; i < 4; i++) acc[i] = vzero;

    const bool has_bn = (scale != nullptr);
    int ci_ld = tid & 63;
    float sc = 1.f, sh = 0.f;
    if (has_bn) { sc = scale[ci_ld]; sh = shift[ci_ld]; }

    int arow  = (wave * 16 + lo) * A_STRIDE;
    int bcol  = hi * BP_STRIDE + lo * 2;     // per-lane base within a kpair pair-row

    for (int t = 0; t < 27; t++) {
        int dz = t / 9 - 1;
        int dy = (t / 3) % 3 - 1;
        int dx = t % 3 - 1;
        // ---- load weights for this tap into K-pair-interleaved layout ----
#pragma unroll
        for (int it = 0; it < 16; it++) {
            int j  = tid + it * 256;
            int ci = j >> 6, co = j & 63;
            Blds[(ci >> 1) * BP_STRIDE + co * 2 + (ci & 1)] = w[t * 4096 + j];
        }
        // prefetch next tap's weight slab while we compute this one
        if (t < 26) __builtin_prefetch(&w[(t + 1) * 4096 + (tid << 4)], 0, 0);
        // ---- load A tile: 128 m x 64 ci (im2col for this tap) ----
        int z = z0 + dz;
        bool zok = ((unsigned)z < 32u);
#pragma unroll 4
        for (int it = 0; it < 32; it++) {
            int m  = (tid >> 6) + it * 4;
            int yy = y0 + (m >> 5) + dy;
            int xx = (m & 31) + dx;
            float v = 0.f;
            if (zok && (unsigned)yy < 32u && (unsigned)xx < 32u) {
                v = in[(((size_t)b * R3) + (size_t)(z * 1024 + yy * 32 + xx)) * CH + ci_ld];
                if (has_bn) v = lrelu(sc * v + sh);
            }
            Alds[m * A_STRIDE + ci_ld] = v;
        }
        __syncthreads();
        // ---- 16 K-steps of 4, 4 Cout tiles; 5 ds_load_b64 per 4 WMMAs ----
#pragma unroll
        for (int kk = 0; kk < 16; kk++) {
            int ka = kk * 4 + 2 * hi;                 // even -> 8B aligned
            v2f a = *(const v2f*)(Alds + arow + ka);
            int brow = (kk * 2) * BP_STRIDE + bcol;   // kpair row = kk*2 + hi
#pragma unroll
            for (int ct = 0; ct < 4; ct++) {
                v2f bb = *(const v2f*)(Blds + brow + ct * 32);
                acc[ct] = __builtin_amdgcn_wmma_f32_16x16x4_f32(
                    false, a, false, bb, (short)0, acc[ct], false, false);
            }
        }
        __syncthreads();
    }
    // ---- bias + store (channel-last: coalesced 64B runs per lane-half) ----
    size_t obase = ((size_t)b * R3 + sp0) * CH;
#pragma unroll
    for (int ct = 0; ct < 4; ct++) {
        float bv = bias[ct * 16 + lo];
#pragma unroll
        for (int r = 0; r < 8; r++) {
            int m = wave * 16 + r + 8 * hi;     // D layout: M = r + 8*(lane/16)
            out[obase + (size_t)m * CH + ct * 16 + lo] = acc[ct][r] + bv;
        }
    }
}

// ---------------- per-channel sum/sumsq for [B][S][64] channel-last ----------
__global__ __launch_bounds__(256) void k_stats_cl(const float* __restrict__ y,
                                                  float* __restrict__ gsum,
                                                  float* __restrict__ gsq, int S) {
    __shared__ float ssum[256], ssq[256];
    int chunks = S >> 10;
    int b  = blockIdx.x / chunks;
    int ch = blockIdx.x % chunks;
    int c  = threadIdx.x & 63;
    int sl = threadIdx.x >> 6;
    const float* base = y + ((size_t)b * S + (size_t)ch * 1024) * CH + c;
    float s = 0.f, q = 0.f;
    for (int sp = sl; sp < 1024; sp += 4) {
        float v = base[(size_t)sp * CH];
        s += v; q += v * v;
    }
    ssum[threadIdx.x] = s; ssq[threadIdx.x] = q;
    __syncthreads();
    if (threadIdx.x < 64) {
        float ts = ssum[c] + ssum[c + 64] + ssum[c + 128] + ssum[c + 192];
        float tq = ssq[c] + ssq[c + 64] + ssq[c + 128] + ssq[c + 192];
        atomicAdd(&gsum[c], ts);
        atomicAdd(&gsq[c], tq);
    }
}

// ---------------- per-channel sum/sumsq for [B][64][N] (p) ----------------
__global__ __launch_bounds__(256) void k_stats_row(const float* __restrict__ p,
                                                   float* __restrict__ gsum,
                                                   float* __restrict__ gsq) {
    __shared__ float ssum[256], ssq[256];
    int row = blockIdx.x;                 // b*64 + d
    const float* base = p + (size_t)row * NPTS;
    float s = 0.f, q = 0.f;
    for (int i = threadIdx.x; i < NPTS; i += 256) { float v = base[i]; s += v; q += v * v; }
    ssum[threadIdx.x] = s; ssq[threadIdx.x] = q;
    __syncthreads();
    for (int st = 128; st > 0; st >>= 1) {
        if (threadIdx.x < st) { ssum[threadIdx.x] += ssum[threadIdx.x + st]; ssq[threadIdx.x] += ssq[threadIdx.x + st]; }
        __syncthreads();
    }
    if (threadIdx.x == 0) {
        int d = row & 63;
        atomicAdd(&gsum[d], ssum[0]);
        atomicAdd(&gsq[d], ssq[0]);
    }
}

// ---------------- fold stats into per-channel scale/shift ----------------
__global__ void k_finalize_bn(const float* __restrict__ gsum, const float* __restrict__ gsq,
                              const float* __restrict__ gamma, const float* __restrict__ beta,
                              float count, float* __restrict__ scale, float* __restrict__ shift) {
    int c = threadIdx.x;
    float mean = gsum[c] / count;
    float var  = gsq[c] / count - mean * mean;
    float s    = gamma[c] * rsqrtf(var + BN_EPS);
    scale[c] = s;
    shift[c] = beta[c] - mean * s;
}

// ---------------- point MLP: p[b,d,n] = sum_c f[b,c,n] * W[c,d] + bias[d] ----
__global__ __launch_bounds__(256) void k_point_mlp(const float* __restrict__ feat,
                                                   const float* __restrict__ pw,
                                                   const float* __restrict__ pb,
                                                   float* __restrict__ p) {
    __shared__ float Fl[64 * 64];   // [c][n]
    __shared__ float Wl[64 * 64];   // [c][d]
    int blk = blockIdx.x;
    int b  = blk >> 8;              // 256 n-blocks per batch
    int n0 = (blk & 255) * 64;
#pragma unroll
    for (int it = 0; it < 16; it++) {
        int j = threadIdx.x + it * 256;
        Fl[j] = feat[((size_t)b * CH + (j >> 6)) * NPTS + n0 + (j & 63)];
        Wl[j] = pw[j];
    }
    __syncthreads();
    int n  = threadIdx.x & 63;
    int dq = threadIdx.x >> 6;
    float acc[16];
#pragma unroll
    for (int d = 0; d < 16; d++) acc[d] = 0.f;
    for (int c = 0; c < 64; c++) {
        float fv = Fl[c * 64 + n];
#pragma unroll
        for (int d = 0; d < 16; d++) acc[d] += fv * Wl[c * 64 + dq * 16 + d];
    }
#pragma unroll
    for (int d = 0; d < 16; d++)
        p[((size_t)b * CH + dq * 16 + d) * NPTS + n0 + n] = acc[d] + pb[dq * 16 + d];
}

// ---------------- devoxelize (BN2+LReLU per voxel) + point branch ----------
__global__ __launch_bounds__(256) void k_final(const float* __restrict__ y2,
                                               const float* __restrict__ p,
                                               const float* __restrict__ nrm,
                                               const float* __restrict__ sc2,
                                               const float* __restrict__ sh2,
                                               const float* __restrict__ scp,
                                               const float* __restrict__ shp,
                                               float* __restrict__ out) {
    __shared__ int   cidx[64][8];
    __shared__ float cwgt[64][8];
    __shared__ float s2s[64], t2s[64], sps[64], tps[64];
    int blk = blockIdx.x;
    int b  = blk >> 8;
    int n0 = (blk & 255) * 64;
    int tid = threadIdx.x;
    if (tid < 64) {
        s2s[tid] = sc2[tid]; t2s[tid] = sh2[tid];
        sps[tid] = scp[tid]; tps[tid] = shp[tid];
        int n = n0 + tid;
        int c0[3], c1[3]; float fr[3];
        for (int d = 0; d < 3; d++) {
            float v  = nrm[((size_t)b * 3 + d) * NPTS + n];
            float fl = floorf(v);
            c0[d] = (int)fl;
            c1[d] = min(c0[d] + 1, RR - 1);
            fr[d] = v - fl;
        }
        for (int k = 0; k < 8; k++) {
            int   xs = (k & 4) ? c1[0] : c0[0]; float wx = (k & 4) ? fr[0] : 1.f - fr[0];
            int   ys = (k & 2) ? c1[1] : c0[1]; float wy = (k & 2) ? fr[1] : 1.f - fr[1];
            int   zs = (k & 1) ? c1[2] : c0[2]; float wz = (k & 1) ? fr[2] : 1.f - fr[2];
            cidx[tid][k] = (xs * RR + ys) * RR + zs;
            cwgt[tid][k] = wx * wy * wz;
        }
    }
    __syncthreads();
    int n  = tid & 63;
    int cq = tid >> 6;
    float accs[16];
#pragma unroll
    for (int j = 0; j < 16; j++) {
        int c = cq * 16 + j;
        float pv = p[((size_t)b * CH + c) * NPTS + n0 + n];
        accs[j] = lrelu(sps[c] * pv + tps[c]);
    }
    for (int k = 0; k < 8; k++) {
        const float* gb = y2 + ((size_t)(b * R3 + cidx[n][k])) * CH + cq * 16;
        float wk = cwgt[n][k];
#pragma unroll
        for (int j = 0; j < 16; j++) {
            int c = cq * 16 + j;
            accs[j] += wk * lrelu(s2s[c] * gb[j] + t2s[c]);
        }
    }
#pragma unroll
    for (int j = 0; j < 16; j++)
        out[((size_t)b * CH + cq * 16 + j) * NPTS + n0 + n] = accs[j];
}

extern "C" void kernel_launch(void* const* d_in, const int* in_sizes, int n_in,
                              void* d_out, int out_size, void* d_ws, size_t ws_size,
                              hipStream_t stream) {
    const float* feat   = (const float*)d_in[0];
    const float* coords = (const float*)d_in[1];
    const float* w1  = (const float*)d_in[2];
    const float* b1  = (const float*)d_in[3];
    const float* g1  = (const float*)d_in[4];
    const float* be1 = (const float*)d_in[5];
    const float* w2  = (const float*)d_in[6];
    const float* b2  = (const float*)d_in[7];
    const float* g2  = (const float*)d_in[8];
    const float* be2 = (const float*)d_in[9];
    const float* pw  = (const float*)d_in[10];
    const float* pbi = (const float*)d_in[11];
    const float* pg  = (const float*)d_in[12];
    const float* pbe = (const float*)d_in[13];
    float* out = (float*)d_out;
    float* ws  = (float*)d_ws;

    const size_t GRID_ELEMS = (size_t)BATCH * R3 * CH;   // 16,777,216
    float* buf0  = ws;                       // grid -> (later) y2
    float* buf1  = buf0 + GRID_ELEMS;        // y1 -> (later) p [B,64,N]
    float* cnt   = buf1 + GRID_ELEMS;        // 262,144
    float* nrm   = cnt + (size_t)BATCH * R3; // 393,216
    float* bmean = nrm + (size_t)BATCH * 3 * NPTS;  // 32
    float* bmax  = bmean + 32;               // 32
    float* st1   = bmax + 32;                // 128 (sum|sq)
    float* st2   = st1 + 128;
    float* stp   = st2 + 128;
    float* sc1   = stp + 128; float* sh1 = sc1 + 64;
    float* sc2   = sh1 + 64;  float* sh2 = sc2 + 64;
    float* scp   = sh2 + 64;  float* shp = scp + 64;

    // zero the accumulation buffers (graph-capture safe)
    hipMemsetAsync(buf0, 0, GRID_ELEMS * sizeof(float), stream);
    hipMemsetAsync(cnt, 0, (size_t)BATCH * R3 * sizeof(float), stream);
    hipMemsetAsync(st1, 0, 3 * 128 * sizeof(float), stream);

    // voxelization
    k_coord_mean<<<BATCH, 256, 0, stream>>>(coords, bmean);
    k_coord_maxnorm<<<BATCH, 256, 0, stream>>>(coords, bmean, bmax);
    k_voxelize<<<(BATCH * NPTS) / 256, 256, 0, stream>>>(coords, feat, bmean, bmax, buf0, cnt, nrm);
    k_grid_norm<<<(unsigned)(GRID_ELEMS / 256), 256, 0, stream>>>(buf0, cnt);

    // conv1 (raw input) -> y1 in buf1
    k_conv_wmma<<<BATCH * 256, 256, 0, stream>>>(buf0, buf1, w1, b1, nullptr, nullptr);
    k_stats_cl<<<BATCH * (R3 >> 10), 256, 0, stream>>>(buf1, st1, st1 + 64, R3);
    k_finalize_bn<<<1, 64, 0, stream>>>(st1, st1 + 64, g1, be1, (float)(BATCH * R3), sc1, sh1);

    // conv2 with fused BN1+LReLU on input -> y2 in buf0
    k_conv_wmma<<<BATCH * 256, 256, 0, stream>>>(buf1, buf0, w2, b2, sc1, sh1);
    k_stats_cl<<<BATCH * (R3 >> 10), 256, 0, stream>>>(buf0, st2, st2 + 64, R3);
    k_finalize_bn<<<1, 64, 0, stream>>>(st2, st2 + 64, g2, be2, (float)(BATCH * R3), sc2, sh2);

    // point branch: p into buf1 (y1 is dead now)
    k_point_mlp<<<BATCH * (NPTS / 64), 256, 0, stream>>>(feat, pw, pbi, buf1);
    k_stats_row<<<BATCH * CH, 256, 0, stream>>>(buf1, stp, stp + 64);
    k_finalize_bn<<<1, 64, 0, stream>>>(stp, stp + 64, pg, pbe, (float)(BATCH * NPTS), scp, shp);

    // fused devoxelize + point branch -> out[B,C,N]
    k_final<<<BATCH * (NPTS / 64), 256, 0, stream>>>(buf0, buf1, nrm, sc2, sh2, scp, shp, out);

    // second tuple element: coords pass-through
    hipMemcpyAsync(out + (size_t)BATCH * CH * NPTS, coords,
                   (size_t)BATCH * 3 * NPTS * sizeof(float),
                   hipMemcpyDeviceToDevice, stream);
}